// AGNN_72241349918727
// MI455X (gfx1250) — compile-verified
//
#include <hip/hip_runtime.h>
#include <math.h>

// ---------------------------------------------------------------------------
// AGNN forward for MI455X (gfx1250, wave32).
//   x    = relu(X @ W1 + b1)            -> V_WMMA_F32_16X16X4_F32 tiles,
//                                          A staged via GLOBAL_LOAD_ASYNC_TO_LDS
//   x'   = AGNNConv(x, beta, edges)     -> 3-pass segment softmax w/ atomics
//   out  = log_softmax(x' @ W2 + b2)
// ---------------------------------------------------------------------------

typedef __attribute__((ext_vector_type(2))) float v2f;
typedef __attribute__((ext_vector_type(8))) float v8f;

#define NFEAT 100
#define HID   32

// ---------------------------------------------------------------------------
// 0) init: m = -inf, denom = 0, agg = 0  (workspace is poisoned, not zeroed)
// ---------------------------------------------------------------------------
__global__ __launch_bounds__(256)
void agnn_init(float* __restrict__ m, float* __restrict__ denom,
               float* __restrict__ agg, int N) {
  int i = blockIdx.x * blockDim.x + threadIdx.x;
  if (i < N * HID) agg[i] = 0.0f;
  if (i < N) { m[i] = -INFINITY; denom[i] = 0.0f; }
}

// ---------------------------------------------------------------------------
// 1) GEMM1 + bias + ReLU + per-row inverse norm.
//    One wave computes a 16x32 tile: 2 f32 accumulators, 25 K-steps of
//    V_WMMA_F32_16X16X4_F32. 4 waves / 128-thread block.
//    ISA layouts (CDNA5 ISA):
//      A 16x4 f32 : lanes 0-15 hold {K0,K1}, lanes 16-31 hold {K2,K3}
//      B 4x16 f32 : mirrored K split, N striped across 16 lanes
//      C/D 16x16  : VGPR r -> row r (lanes 0-15) / row r+8 (lanes 16-31)
//    W1 is staged TRANSPOSED in LDS so each B fragment is one aligned b64.
//    The 6400B A tile is staged with async global->LDS loads (ASYNCcnt).
// ---------------------------------------------------------------------------
__global__ __launch_bounds__(128)
void gemm_relu_norm(const float* __restrict__ X, const float* __restrict__ W1,
                    const float* __restrict__ b1, float* __restrict__ xout,
                    float* __restrict__ inv, int Mtiles, int N) {
  __shared__ float sWt[HID * NFEAT];      // 12.8 KB, W1 transposed: sWt[c*100+k]
  __shared__ float sA[4][16 * NFEAT];     // 4 waves * 6.4 KB A tiles

  const int tid  = threadIdx.x;
  const int wave = tid >> 5;
  const int lane = tid & 31;
  const int tile = blockIdx.x * 4 + wave;

  // stage W1 transposed (block-cooperative, coalesced reads)
  for (int i = tid; i < NFEAT * HID; i += 128) {
    const int k = i >> 5, c = i & 31;     // W1 is [100][32] row-major
    sWt[c * NFEAT + k] = W1[i];
  }

  // stage this wave's A tile: rows of X are contiguous -> contiguous 1600 floats
  const size_t lim = (size_t)N * NFEAT;
  if (tile < Mtiles) {
    const size_t base = (size_t)tile * 16 * NFEAT;
    if (base + 16 * NFEAT <= lim) {
      // fast path: async copy 6400 B global -> LDS, 25 x b64 per lane
      const unsigned long long gbase =
          (unsigned long long)(const void*)(X + base) + (unsigned)(lane * 8);
      const unsigned lbase =
          (unsigned)(uintptr_t)(&sA[wave][0]) + (unsigned)(lane * 8);
#pragma unroll
      for (int i = 0; i < 25; ++i) {
        const unsigned       loff = lbase + (unsigned)(i * 256);
        const unsigned long long ga = gbase + (unsigned long long)(i * 256);
        asm volatile("global_load_async_to_lds_b64 %0, %1, off"
                     :: "v"(loff), "v"(ga) : "memory");
      }
    } else {
      for (int i = lane; i < 16 * NFEAT; i += 32)
        sA[wave][i] = (base + i < lim) ? X[base + i] : 0.0f;
    }
  }
  __syncthreads();                       // W1 visibility across waves
  if (tile >= Mtiles) return;            // whole-wave exit: EXEC stays all-1s
  asm volatile("s_wait_asynccnt 0x0" ::: "memory");  // own A tile landed

  const int half = lane >> 4;            // 0: K even pair / rows r ; 1: K odd pair / rows r+8
  const int l16  = lane & 15;
  const float* As = sA[wave];

  v8f c0 = {};   // cols 0..15
  v8f c1 = {};   // cols 16..31

#pragma unroll
  for (int kb = 0; kb < NFEAT; kb += 4) {
    const int k2 = kb + 2 * half;        // even -> 8B-aligned pairs everywhere
    const v2f a   = *(const v2f*)&As [l16        * NFEAT + k2];
    const v2f b0  = *(const v2f*)&sWt[l16        * NFEAT + k2];
    const v2f b1v = *(const v2f*)&sWt[(16 + l16) * NFEAT + k2];
    c0 = __builtin_amdgcn_wmma_f32_16x16x4_f32(false, a, false, b0,  (short)0, c0, false, false);
    c1 = __builtin_amdgcn_wmma_f32_16x16x4_f32(false, a, false, b1v, (short)0, c1, false, false);
  }

  // epilogue: bias + relu + store x; fused row-norm reduction
  const float bias0 = b1[l16];
  const float bias1 = b1[16 + l16];
  float r0[8], r1[8];
#pragma unroll
  for (int r = 0; r < 8; ++r) {
    const int row = tile * 16 + r + 8 * half;
    float v0 = fmaxf(c0[r] + bias0, 0.0f);
    float v1 = fmaxf(c1[r] + bias1, 0.0f);
    r0[r] = v0; r1[r] = v1;
    if (row < N) {
      xout[(size_t)row * HID + l16]      = v0;
      xout[(size_t)row * HID + 16 + l16] = v1;
    }
  }
#pragma unroll
  for (int r = 0; r < 8; ++r) {
    float s = r0[r] * r0[r] + r1[r] * r1[r];
    s += __shfl_xor(s, 8, 16);   // reduce over the 16-lane half holding this row
    s += __shfl_xor(s, 4, 16);
    s += __shfl_xor(s, 2, 16);
    s += __shfl_xor(s, 1, 16);
    const int row = tile * 16 + r + 8 * half;
    if (l16 == r && row < N)
      inv[row] = 1.0f / fmaxf(sqrtf(s), 1e-12f);
  }
}

// ---------------------------------------------------------------------------
// 2) Pass A: alpha_e = beta * <x_s,x_d> * inv_s * inv_d ; segment max over dst
//    One wave per edge; lane = feature. Coalesced 128B gathers (L2-resident).
// ---------------------------------------------------------------------------
__global__ __launch_bounds__(256)
void edge_alpha_max(const float* __restrict__ x, const float* __restrict__ inv,
                    const int* __restrict__ src, const int* __restrict__ dst,
                    const float* __restrict__ beta_p, float* __restrict__ alpha,
                    float* __restrict__ m, int E) {
  const int lane = threadIdx.x & 31;
  const int e = (blockIdx.x * blockDim.x + threadIdx.x) >> 5;
  if (e >= E) return;
  const int s = src[e], d = dst[e];
  float p = x[(size_t)s * HID + lane] * x[(size_t)d * HID + lane];
  p += __shfl_xor(p, 16);
  p += __shfl_xor(p, 8);
  p += __shfl_xor(p, 4);
  p += __shfl_xor(p, 2);
  p += __shfl_xor(p, 1);
  if (lane == 0) {
    const float a = beta_p[0] * p * inv[s] * inv[d];
    alpha[e] = a;
    atomicMax(&m[d], a);               // global_atomic_max_num_f32
  }
}

// ---------------------------------------------------------------------------
// 3) Pass B: a_e = exp(alpha_e - m[dst]); denom[dst] += a_e   (in-place alpha)
// ---------------------------------------------------------------------------
__global__ __launch_bounds__(256)
void edge_exp_sum(float* __restrict__ alpha, const int* __restrict__ dst,
                  const float* __restrict__ m, float* __restrict__ denom, int E) {
  const int e = blockIdx.x * blockDim.x + threadIdx.x;
  if (e >= E) return;
  const int d = dst[e];
  const float a = expf(alpha[e] - m[d]);
  alpha[e] = a;
  atomicAdd(&denom[d], a);
}

// ---------------------------------------------------------------------------
// 4) Pass C: agg[dst] += (a_e / denom[dst]) * x[src]   (wave per edge)
// ---------------------------------------------------------------------------
__global__ __launch_bounds__(256)
void edge_aggregate(const float* __restrict__ x, const float* __restrict__ alpha,
                    const int* __restrict__ src, const int* __restrict__ dst,
                    const float* __restrict__ denom, float* __restrict__ agg, int E) {
  const int lane = threadIdx.x & 31;
  const int e = (blockIdx.x * blockDim.x + threadIdx.x) >> 5;
  if (e >= E) return;
  const int s = src[e], d = dst[e];
  const float w = alpha[e] / denom[d];
  atomicAdd(&agg[(size_t)d * HID + lane], w * x[(size_t)s * HID + lane]);
}

// ---------------------------------------------------------------------------
// 5) Head: logits = agg @ W2 + b2 ; log_softmax over 2 classes
// ---------------------------------------------------------------------------
__global__ __launch_bounds__(256)
void head_logsoftmax(const float* __restrict__ agg, const float* __restrict__ W2,
                     const float* __restrict__ b2, float* __restrict__ out, int N) {
  const int i = blockIdx.x * blockDim.x + threadIdx.x;
  if (i >= N) return;
  const float* h = agg + (size_t)i * HID;
  float l0 = b2[0], l1 = b2[1];
#pragma unroll
  for (int j = 0; j < HID; ++j) {
    const float v = h[j];
    l0 = fmaf(v, W2[j * 2 + 0], l0);
    l1 = fmaf(v, W2[j * 2 + 1], l1);
  }
  const float mx  = fmaxf(l0, l1);
  const float lse = mx + logf(expf(l0 - mx) + expf(l1 - mx));
  out[(size_t)i * 2 + 0] = l0 - lse;
  out[(size_t)i * 2 + 1] = l1 - lse;
}

// ---------------------------------------------------------------------------
extern "C" void kernel_launch(void* const* d_in, const int* in_sizes, int n_in,
                              void* d_out, int out_size, void* d_ws, size_t ws_size,
                              hipStream_t stream) {
  const float* X    = (const float*)d_in[0];
  const float* W1   = (const float*)d_in[1];
  const float* b1   = (const float*)d_in[2];
  const float* beta = (const float*)d_in[3];
  const float* W2   = (const float*)d_in[4];
  const float* b2   = (const float*)d_in[5];
  const int*   ei   = (const int*)d_in[6];

  const int N = in_sizes[0] / NFEAT;       // 100000
  const int E = in_sizes[6] / 2;           // 3,300,000
  const int* src = ei;
  const int* dst = ei + E;

  // workspace layout (floats): x[N*32] | inv[N] | m[N] | denom[N] | alpha[E] | agg[N*32]
  float* ws    = (float*)d_ws;
  float* x     = ws;
  float* inv   = x     + (size_t)N * HID;
  float* mbuf  = inv   + N;
  float* den   = mbuf  + N;
  float* alpha = den   + N;
  float* agg   = alpha + E;

  float* out = (float*)d_out;

  const int Ntot = N * HID;
  agnn_init<<<(Ntot + 255) / 256, 256, 0, stream>>>(mbuf, den, agg, N);

  const int Mtiles = (N + 15) / 16;
  gemm_relu_norm<<<(Mtiles + 3) / 4, 128, 0, stream>>>(X, W1, b1, x, inv, Mtiles, N);

  const long ethreads = (long)E * 32;
  edge_alpha_max<<<(int)((ethreads + 255) / 256), 256, 0, stream>>>(
      x, inv, src, dst, beta, alpha, mbuf, E);
  edge_exp_sum<<<(E + 255) / 256, 256, 0, stream>>>(alpha, dst, mbuf, den, E);
  edge_aggregate<<<(int)((ethreads + 255) / 256), 256, 0, stream>>>(
      x, alpha, src, dst, den, agg, E);

  head_logsoftmax<<<(N + 255) / 256, 256, 0, stream>>>(agg, W2, b2, out, N);
}